// YOLOV3DynamicTargetGeneratorSimple_59227599012160
// MI455X (gfx1250) — compile-verified
//
#include <hip/hip_runtime.h>
#include <stdint.h>

typedef float v4f __attribute__((ext_vector_type(4)));

#define TPB 256
#define NUM_GT 50

// ---------------------------------------------------------------------------
// Fill kernel: writes the constant part of the output
//   [BN .. 7*BN)   : center/scale/weight  -> 0.0f   (6*BN floats)
//   [7*BN .. 87*BN): class_t              -> -1.0f  (80*BN floats)
// One float4 non-temporal store per thread (253 MB stream; NT keeps it out of
// L2 since it exceeds the 192 MB global L2 anyway).
// ---------------------------------------------------------------------------
__global__ __launch_bounds__(TPB) void yolo_fill_kernel(float* __restrict__ out,
                                                        long long zeros4,
                                                        long long total4) {
  long long i = (long long)blockIdx.x * TPB + threadIdx.x;
  if (i >= total4) return;
  float val = (i < zeros4) ? 0.0f : -1.0f;
  v4f v = {val, val, val, val};
  __builtin_nontemporal_store(v, ((v4f*)out) + i);
}

// ---------------------------------------------------------------------------
// Objness kernel: one thread per (b, n). The 50 gt boxes (800 B) of batch b
// are DMA'd into LDS with the gfx1250 async-to-LDS path (ASYNCcnt), then each
// thread runs a fully unrolled max-IoU loop against LDS broadcasts.
// IoU > 0.5  <=>  inter > 0.5 * max(union, EPS)   (no division needed).
// ---------------------------------------------------------------------------
__global__ __launch_bounds__(TPB) void yolo_objness_kernel(
    const v4f* __restrict__ box_preds, const float* __restrict__ gt_boxes,
    float* __restrict__ obj, int N) {
  __shared__ float s_gt[NUM_GT * 4];
  __shared__ float s_area[NUM_GT];

  const int b = blockIdx.y;
  const int tid = threadIdx.x;
  const int n = blockIdx.x * TPB + tid;

  // Lanes 0..49 (waves 0 and 1) each async-copy one 16B gt box into LDS.
  if (tid < NUM_GT) {
    const float* src = gt_boxes + ((size_t)b * NUM_GT + tid) * 4;
    unsigned lds_off = (unsigned)(size_t)(&s_gt[tid * 4]);  // flat->LDS offset
    asm volatile("global_load_async_to_lds_b128 %0, %1, off"
                 :
                 : "v"(lds_off), "v"(src)
                 : "memory");
  }
  asm volatile("s_wait_asynccnt 0" ::: "memory");
  __syncthreads();

  // Precompute gt areas once per block.
  if (tid < NUM_GT) {
    v4f g = ((const v4f*)s_gt)[tid];
    s_area[tid] = (g.z - g.x) * (g.w - g.y);
  }
  __syncthreads();

  if (n < N) {
    v4f p = box_preds[(size_t)b * N + n];  // global_load_b128
    float a1 = (p.z - p.x) * (p.w - p.y);
    float best = -1.0f;  // holds max(inter - 0.5*union'); >0 <=> IoU > 0.5
#pragma unroll
    for (int m = 0; m < NUM_GT; ++m) {
      v4f g = ((const v4f*)s_gt)[m];  // ds_load_b128, same-addr broadcast
      float tlx = fmaxf(p.x, g.x);
      float tly = fmaxf(p.y, g.y);
      float brx = fminf(p.z, g.z);
      float bry = fminf(p.w, g.w);
      float w = fmaxf(brx - tlx, 0.0f);
      float h = fmaxf(bry - tly, 0.0f);
      float inter = w * h;
      float uni = fmaxf(a1 + s_area[m] - inter, 1e-6f);
      best = fmaxf(best, fmaf(-0.5f, uni, inter));
    }
    __builtin_nontemporal_store(best > 0.0f ? -1.0f : 0.0f,
                                obj + (size_t)b * N + n);
  }
}

// ---------------------------------------------------------------------------
extern "C" void kernel_launch(void* const* d_in, const int* in_sizes, int n_in,
                              void* d_out, int out_size, void* d_ws,
                              size_t ws_size, hipStream_t stream) {
  const v4f* box_preds = (const v4f*)d_in[0];   // [B, N, 4] f32
  const float* gt_boxes = (const float*)d_in[1];  // [B, 50, 4] f32
  float* out = (float*)d_out;  // objness | center | scale | weight | class

  const int B = in_sizes[1] / (NUM_GT * 4);     // 6400 / 200 = 32
  const int N = in_sizes[0] / (4 * B);          // 22743
  const long long BN = (long long)B * N;        // 727776, multiple of 4

  // Constant region: floats [BN, 87*BN). First 6*BN are zeros, rest -1.
  const long long zeros4 = (6 * BN) / 4;
  const long long total4 = (86 * BN) / 4;
  const int fill_blocks = (int)((total4 + TPB - 1) / TPB);
  yolo_fill_kernel<<<fill_blocks, TPB, 0, stream>>>(out + BN, zeros4, total4);

  dim3 grid((N + TPB - 1) / TPB, B);
  yolo_objness_kernel<<<grid, TPB, 0, stream>>>(box_preds, gt_boxes, out, N);
}